// GaussianMixtureModel_80831284510831
// MI455X (gfx1250) — compile-verified
//
#include <hip/hip_runtime.h>

// ---------------- problem constants ----------------
#define NPIX     (1u << 20)     // N
#define NCOMP    16             // M*G mixture components
#define FSZ      12             // feature row stride (10 real + 2 pad for K%4==0)
#define NFEAT    10
#define TPB      256
#define NBLK     512
#define WPB      (TPB / 32)
#define NWAVES   (NBLK * WPB)
#define NTILES   (NPIX / 16)    // 16 pixels per wave-tile
#define NSLOT    32             // global accumulation replicas
#define STATSZ   (NCOMP * FSZ)  // 192
#define EPSV     1e-4f
#define C_LOG2PI 5.51363122f    // 3 * log(2*pi)

typedef __attribute__((ext_vector_type(2))) float v2f;
typedef __attribute__((ext_vector_type(8))) float v8f;

// logp(16 comps x 16 pixels) = Coef(16x12) * Feat(12x16) via 3x V_WMMA_F32_16X16X4_F32
__device__ __forceinline__ v8f wmma_logp(const v2f A[3], const v2f B[3]) {
    v8f acc = {0.f, 0.f, 0.f, 0.f, 0.f, 0.f, 0.f, 0.f};
    acc = __builtin_amdgcn_wmma_f32_16x16x4_f32(false, A[0], false, B[0], (short)0, acc, false, false);
    acc = __builtin_amdgcn_wmma_f32_16x16x4_f32(false, A[1], false, B[1], (short)0, acc, false, false);
    acc = __builtin_amdgcn_wmma_f32_16x16x4_f32(false, A[2], false, B[2], (short)0, acc, false, false);
    return acc;
}

// A operand (16x4 f32): VGPR0 = K0 (lanes 0-15) / K2 (lanes 16-31), VGPR1 = K1/K3.
__device__ __forceinline__ void load_coef_A(const float* __restrict__ coef,
                                            int row, int h, v2f A[3]) {
#pragma unroll
    for (int j = 0; j < 3; ++j) {
        const float* cr = coef + row * FSZ + 4 * j + 2 * h;
        A[j].x = cr[0];
        A[j].y = cr[1];
    }
}

__global__ void gmm_zero(float* __restrict__ slots) {
    int i = blockIdx.x * blockDim.x + threadIdx.x;
    if (i < NSLOT * STATSZ) slots[i] = 0.0f;
}

// M-step statistics pass.  mode 0: init responsibilities (g == n%4 of labeled
// mixture).  mode 1: WMMA E-step with previous round's coefficients, softmax
// over the labeled mixture's 4 gaussians, accumulate resp-weighted features.
__global__ __launch_bounds__(TPB) void gmm_stats(const float* __restrict__ x,
                                                 const int* __restrict__ label,
                                                 const float* __restrict__ coef,
                                                 float* __restrict__ slots,
                                                 int mode) {
    __shared__ float lstat[4 * STATSZ];
    for (int i = threadIdx.x; i < 4 * STATSZ; i += TPB) lstat[i] = 0.0f;
    __syncthreads();

    const int lane = threadIdx.x & 31;
    const int col  = lane & 15;   // pixel-in-tile, also coef row for A operand
    const int h    = lane >> 4;   // half-wave (K-split) selector
    const int rep  = lane & 3;    // LDS replica to spread ds_add collisions

    v2f A[3] = {};
    if (mode) load_coef_A(coef, col, h, A);

    const int wave0 = blockIdx.x * WPB + (threadIdx.x >> 5);
    for (int t = wave0; t < NTILES; t += NWAVES) {
        const int p = (t << 4) + col;
        __builtin_prefetch(x + p + NWAVES * 16, 0, 1);          // global_prefetch_b8
        const float x0 = x[p];
        const float x1 = x[p + NPIX];
        const float x2 = x[p + 2 * NPIX];
        const int   lab = label[p];

        float f[NFEAT];
        f[0] = 1.0f;    f[1] = x0;      f[2] = x1;      f[3] = x2;
        f[4] = x0 * x0; f[5] = x0 * x1; f[6] = x0 * x2;
        f[7] = x1 * x1; f[8] = x1 * x2; f[9] = x2 * x2;

        const bool active = (h == (lab >> 1));  // exactly one half-lane owns the labeled mixture
        float* const dst = &lstat[rep * STATSZ + (lab << 2) * FSZ];

        if (mode == 0) {
            if (active) {
                const int g = p & 3;            // round-robin init assignment
#pragma unroll
                for (int k = 0; k < NFEAT; ++k) atomicAdd(dst + g * FSZ + k, f[k]);
            }
        } else {
            // B operand (12x16 feats x pixels), K-split matching A
            v2f B[3];
            B[0].x = h ? x1   : 1.0f;  B[0].y = h ? x2   : x0;
            B[1].x = h ? f[6] : f[4];  B[1].y = h ? f[7] : f[5];
            B[2].x = h ? 0.f  : f[8];  B[2].y = h ? 0.f  : f[9];
            v8f lp = wmma_logp(A, B);   // lane holds comps 8h..8h+7 of pixel `col`
            if (active) {
                const bool hi = (lab & 1);      // which of this lane's two mixtures
                float l0 = hi ? lp[4] : lp[0];
                float l1 = hi ? lp[5] : lp[1];
                float l2 = hi ? lp[6] : lp[2];
                float l3 = hi ? lp[7] : lp[3];
                float mx = fmaxf(fmaxf(l0, l1), fmaxf(l2, l3));
                float e0 = __expf(l0 - mx), e1 = __expf(l1 - mx);
                float e2 = __expf(l2 - mx), e3 = __expf(l3 - mx);
                // hardware v_rcp_f32: softmax normalizer, ~1ulp is plenty
                float inv = __builtin_amdgcn_rcpf(e0 + e1 + e2 + e3);
                float r[4] = {e0 * inv, e1 * inv, e2 * inv, e3 * inv};
#pragma unroll
                for (int g = 0; g < 4; ++g)
#pragma unroll
                    for (int k = 0; k < NFEAT; ++k)
                        atomicAdd(dst + g * FSZ + k, r[g] * f[k]);
            }
        }
    }

    __syncthreads();
    float* const slot = slots + (blockIdx.x & (NSLOT - 1)) * STATSZ;
    for (int i = threadIdx.x; i < STATSZ; i += TPB) {
        float s = lstat[i] + lstat[STATSZ + i] + lstat[2 * STATSZ + i] + lstat[3 * STATSZ + i];
        atomicAdd(slot + i, s);                 // global_atomic_add_f32, 16-way contention
    }
}

// Tiny solve: stats -> (mu, cov, prec=adj/det, logdet, pi) -> 16x12 coef matrix.
// Exact IEEE divides kept here: runs once per round on 16 lanes.
__global__ __launch_bounds__(256) void gmm_params(const float* __restrict__ slots,
                                                  float* __restrict__ coef) {
    __shared__ float S[STATSZ];
    __shared__ float wbuf[NCOMP];
    const int t = threadIdx.x;
    if (t < STATSZ) {
        float s = 0.0f;
        for (int b = 0; b < NSLOT; ++b) s += slots[b * STATSZ + t];
        S[t] = s;
    }
    __syncthreads();
    if (t < NCOMP) wbuf[t] = S[t * FSZ] + EPSV;
    __syncthreads();
    if (t < NCOMP) {
        const float* s = &S[t * FSZ];
        const int m = t >> 2;
        const float w    = wbuf[t];
        const float wsum = wbuf[4 * m] + wbuf[4 * m + 1] + wbuf[4 * m + 2] + wbuf[4 * m + 3];
        const float mu0 = s[1] / w, mu1 = s[2] / w, mu2 = s[3] / w;
        const float c00 = s[4] / w - mu0 * mu0 + EPSV;
        const float c01 = s[5] / w - mu0 * mu1;
        const float c02 = s[6] / w - mu0 * mu2;
        const float c11 = s[7] / w - mu1 * mu1 + EPSV;
        const float c12 = s[8] / w - mu1 * mu2;
        const float c22 = s[9] / w - mu2 * mu2 + EPSV;
        const float d00 = c11 * c22 - c12 * c12;
        const float d01 = c02 * c12 - c01 * c22;
        const float d02 = c01 * c12 - c02 * c11;
        const float det = c00 * d00 + c01 * d01 + c02 * d02;
        const float idt = 1.0f / det;
        const float P00 = d00 * idt, P01 = d01 * idt, P02 = d02 * idt;
        const float P11 = (c00 * c22 - c02 * c02) * idt;
        const float P12 = (c01 * c02 - c00 * c12) * idt;
        const float P22 = (c00 * c11 - c01 * c01) * idt;
        const float Pm0 = P00 * mu0 + P01 * mu1 + P02 * mu2;
        const float Pm1 = P01 * mu0 + P11 * mu1 + P12 * mu2;
        const float Pm2 = P02 * mu0 + P12 * mu1 + P22 * mu2;
        const float mPm = mu0 * Pm0 + mu1 * Pm1 + mu2 * Pm2;
        const float cst = __logf(w / wsum) - 0.5f * (__logf(det) + mPm + C_LOG2PI);
        float* o = &coef[t * FSZ];
        o[0] = cst;  o[1] = Pm0;  o[2] = Pm1;  o[3] = Pm2;
        o[4] = -0.5f * P00;  o[5] = -P01;  o[6] = -P02;
        o[7] = -0.5f * P11;  o[8] = -P12;  o[9] = -0.5f * P22;
        o[10] = 0.0f;        o[11] = 0.0f;
    }
}

// Final E-step + posterior over mixtures, [4, N] output.
__global__ __launch_bounds__(TPB) void gmm_post(const float* __restrict__ x,
                                                const float* __restrict__ coef,
                                                float* __restrict__ out) {
    const int lane = threadIdx.x & 31;
    const int col  = lane & 15;
    const int h    = lane >> 4;
    v2f A[3];
    load_coef_A(coef, col, h, A);

    const int wave0 = blockIdx.x * WPB + (threadIdx.x >> 5);
    for (int t = wave0; t < NTILES; t += NWAVES) {
        const int p = (t << 4) + col;
        __builtin_prefetch(x + p + NWAVES * 16, 0, 1);
        const float x0 = x[p];
        const float x1 = x[p + NPIX];
        const float x2 = x[p + 2 * NPIX];
        v2f B[3];
        B[0].x = h ? x1      : 1.0f;     B[0].y = h ? x2      : x0;
        B[1].x = h ? x0 * x2 : x0 * x0;  B[1].y = h ? x1 * x1 : x0 * x1;
        B[2].x = h ? 0.f     : x1 * x2;  B[2].y = h ? 0.f     : x2 * x2;
        v8f lp = wmma_logp(A, B);        // lane: mixtures (2h, 2h+1) of pixel col
        float ma = fmaxf(fmaxf(lp[0], lp[1]), fmaxf(lp[2], lp[3]));
        float La = ma + __logf(__expf(lp[0] - ma) + __expf(lp[1] - ma) +
                               __expf(lp[2] - ma) + __expf(lp[3] - ma));
        float mb = fmaxf(fmaxf(lp[4], lp[5]), fmaxf(lp[6], lp[7]));
        float Lb = mb + __logf(__expf(lp[4] - mb) + __expf(lp[5] - mb) +
                               __expf(lp[6] - mb) + __expf(lp[7] - mb));
        // softmax over 4 mixtures: other two live in the opposite half-lane
        float hm = fmaxf(La, Lb);
        float gm = fmaxf(hm, __shfl_xor(hm, 16, 32));
        float ea = __expf(La - gm), eb = __expf(Lb - gm);
        float s  = ea + eb;
        float inv = __builtin_amdgcn_rcpf(s + __shfl_xor(s, 16, 32));
        out[(2 * h) * NPIX + p]     = ea * inv;
        out[(2 * h + 1) * NPIX + p] = eb * inv;
    }
}

extern "C" void kernel_launch(void* const* d_in, const int* in_sizes, int n_in,
                              void* d_out, int out_size, void* d_ws, size_t ws_size,
                              hipStream_t stream) {
    (void)in_sizes; (void)n_in; (void)out_size; (void)ws_size;
    const float* x     = (const float*)d_in[0];  // [1,3,N] -> three N-planes
    const int*   label = (const int*)d_in[1];    // [1,N]
    float*       out   = (float*)d_out;          // [1,4,N]
    float*       coef  = (float*)d_ws;           // 16x12 coefficients
    float*       slots = coef + STATSZ;          // 32 x 192 partial stats

    for (int round = 0; round < 3; ++round) {
        gmm_zero  <<<(NSLOT * STATSZ + TPB - 1) / TPB, TPB, 0, stream>>>(slots);
        gmm_stats <<<NBLK, TPB, 0, stream>>>(x, label, coef, slots, round == 0 ? 0 : 1);
        gmm_params<<<1, 256, 0, stream>>>(slots, coef);
    }
    gmm_post<<<NBLK, TPB, 0, stream>>>(x, coef, out);
}